// L2DEncoder_37400575214207
// MI455X (gfx1250) — compile-verified
//
#include <hip/hip_runtime.h>
#include <hip/hip_bf16.h>

// ---------------------------------------------------------------------------
// L2D GNN encoder for MI455X (gfx1250, wave32).
//
//  * adj_prec einsum  -> pure shift (zero FLOPs instead of 8.2 GMAC/layer)
//  * adj_mach einsum  -> 20-way segmented sum S[b,v], then S[b,mo[i]]-h[i]
//  * dense GEMMs on the matrix pipe via v_wmma_f32_16x16x4_f32 (fp32)
//  * 32 output rows per block, 2 M-subtiles per wave -> each weight (B)
//    fragment feeds TWO wmma ops: ~1 global load per wmma instead of 2
//  * msg GEMM synthesizes [h | shift(h) | agg] A-tile directly in LDS
//  * FFN fused through LDS (no 131MB intermediate in HBM)
//
// Input flattening assumption: d_in = { proc_time, machine_order,
// next_op_idx, finished_jobs(int32), W_init, b_init,
// then per layer l: W1, W2, Wm, Wp, Ws, b, b1, b2 }  (jax sorted-key order).
// ---------------------------------------------------------------------------

#define BB    64
#define JJ    50
#define MM    20
#define NND   1000           // JJ*MM nodes per batch
#define EE    128
#define FFD   512
#define ROWS  (BB*NND)       // 64000
#define TILES32 (ROWS/32)    // 2000
#define HTOT  (ROWS*EE)      // 8,192,000
#define JOBOUT (BB*JJ*EE)    // 409,600
#define OUTTOT (JOBOUT+HTOT)

typedef __attribute__((ext_vector_type(2))) float v2f;
typedef __attribute__((ext_vector_type(8))) float v8f;

__device__ __forceinline__ v8f wmma4(v2f a, v2f b, v8f c) {
  // D(16x16,f32) = A(16x4,f32) * B(4x16,f32) + C
  return __builtin_amdgcn_wmma_f32_16x16x4_f32(
      /*neg_a=*/false, a, /*neg_b=*/false, b,
      /*c_mod=*/(short)0, c, /*reuse_a=*/false, /*reuse_b=*/false);
}

// --------------------------------------------------------------------------
// h0[row, e] = proc_time[row] * W_init[e] + b_init[e]
// --------------------------------------------------------------------------
__global__ __launch_bounds__(256) void init_h_kernel(
    const float* __restrict__ proc, const float* __restrict__ Wi,
    const float* __restrict__ bi, float* __restrict__ h) {
  int idx = blockIdx.x * 256 + threadIdx.x;
  if (idx >= HTOT) return;
  int e = idx & (EE - 1);
  int row = idx >> 7;
  h[idx] = proc[row] * Wi[e] + bi[e];
}

// --------------------------------------------------------------------------
// S[b, v, e] = sum over nodes j of batch b with mo[b,j]==v of h[b,j,e]
// 512 threads = 4 groups x 128 e-lanes, private LDS copies, reduced at end.
// --------------------------------------------------------------------------
__global__ __launch_bounds__(512) void machine_sums_kernel(
    const float* __restrict__ h, const int* __restrict__ mo,
    float* __restrict__ S) {
  __shared__ float Sl[4 * MM * EE];     // 40 KB
  int b = blockIdx.x;
  int tid = threadIdx.x;
  int e = tid & (EE - 1);
  int g = tid >> 7;                     // 0..3
  for (int i = tid; i < 4 * MM * EE; i += 512) Sl[i] = 0.f;
  __syncthreads();
  const float* hb = h + (size_t)b * NND * EE;
  const int*   mb = mo + b * NND;
  float* Sg = Sl + g * MM * EE;
  int j0 = g * (NND / 4), j1 = j0 + (NND / 4);
  for (int j = j0; j < j1; ++j) {
    int v = mb[j];
    Sg[v * EE + e] += hb[j * EE + e];   // distinct (v,e) per thread: race-free
  }
  __syncthreads();
  for (int i = tid; i < MM * EE; i += 512) {
    S[b * MM * EE + i] =
        Sl[i] + Sl[MM * EE + i] + Sl[2 * MM * EE + i] + Sl[3 * MM * EE + i];
  }
}

// --------------------------------------------------------------------------
// msg = relu( [h | shift(h) | S[mo]-h] @ [Ws;Wp;Wm] + b )
// Block: 32 rows x 128 cols, 8 waves; each wave owns one 16-col N tile and
// TWO 16-row M subtiles -> every B fragment is reused by two wmma ops.
// --------------------------------------------------------------------------
__global__ __launch_bounds__(256) void gemm_msg_kernel(
    const float* __restrict__ h, const float* __restrict__ S,
    const int* __restrict__ mo,
    const float* __restrict__ Ws, const float* __restrict__ Wp,
    const float* __restrict__ Wm, const float* __restrict__ bias,
    float* __restrict__ msg) {
  constexpr int AS = 384 + 4;           // 388 % 64 = 4 -> conflict-free frags
  __shared__ float At[32 * AS];         // ~48.5 KB
  int row0 = blockIdx.x * 32;
  int tid = threadIdx.x;

  // ---- stage fused A tile (32 x 384) -------------------------------------
  for (int idx = tid; idx < 32 * 384; idx += 256) {
    int tr = idx / 384;
    int c  = idx - tr * 384;
    int row = row0 + tr;
    int i = row % NND;                  // node index within batch
    float v;
    if (c < EE) {                       // plain h
      v = h[row * EE + c];
    } else if (c < 2 * EE) {            // predecessor shift within a job
      int c2 = c - EE;
      v = (((i + 1) % MM) != 0) ? h[(row + 1) * EE + c2] : 0.f;
    } else {                            // machine aggregation minus self
      int c2 = c - 2 * EE;
      int bb = row / NND;
      int mv = mo[row];
      v = S[(bb * MM + mv) * EE + c2] - h[row * EE + c2];
    }
    At[tr * AS + c] = v;
  }
  __syncthreads();

  // ---- WMMA: 3 segments x 32 k-steps x 2 M-subtiles = 192 wmma/wave ------
  int lane = tid & 31;
  int wave = tid >> 5;                  // 0..7 -> N tile
  int m  = lane & 15;
  int kh = lane >> 4;                   // 0/1: K half (A/B) and M half (D)
  int ncol = wave * 16 + m;
  v8f acc0 = {}, acc1 = {};
  const float* Wseg[3] = {Ws, Wp, Wm};
  for (int seg = 0; seg < 3; ++seg) {
    const float* Ar0 = At + m * AS + seg * EE + 2 * kh;
    const float* Ar1 = Ar0 + 16 * AS;
    const float* Wcol = Wseg[seg] + ncol + 2 * kh * EE;
#pragma unroll 8
    for (int k = 0; k < EE; k += 4) {
      v2f bf; bf.x = Wcol[k * EE]; bf.y = Wcol[k * EE + EE];
      v2f a0; a0.x = Ar0[k]; a0.y = Ar0[k + 1];
      v2f a1; a1.x = Ar1[k]; a1.y = Ar1[k + 1];
      acc0 = wmma4(a0, bf, acc0);
      acc1 = wmma4(a1, bf, acc1);
    }
  }
  // ---- epilogue: bias + relu ---------------------------------------------
  float bi = bias[ncol];
  int rb = row0 + kh * 8;
#pragma unroll
  for (int v = 0; v < 8; ++v) {
    float v0 = acc0[v] + bi;
    float v1 = acc1[v] + bi;
    msg[(rb + v) * EE + ncol]      = v0 > 0.f ? v0 : 0.f;
    msg[(rb + 16 + v) * EE + ncol] = v1 > 0.f ? v1 : 0.f;
  }
}

// --------------------------------------------------------------------------
// h_out = msg + relu(msg@W1 + b1) @ W2 + b2, fused through LDS, 32 rows.
// GEMM1: wave computes a 32x64 slice of the 32x512 intermediate (8 accs).
// GEMM2: wave computes one 32x16 tile of the output (K=512, 2 accs).
// Every W1/W2 fragment feeds two wmma ops.
// --------------------------------------------------------------------------
__global__ __launch_bounds__(256) void ffn_kernel(
    const float* __restrict__ msg,
    const float* __restrict__ W1, const float* __restrict__ b1,
    const float* __restrict__ W2, const float* __restrict__ b2,
    float* __restrict__ hout) {
  constexpr int MS = EE + 4;            // 132 % 64 = 4
  constexpr int FS = FFD + 4;           // 516 % 64 = 4
  __shared__ float Mt[32 * MS];         // msg tile   (~16.9 KB)
  __shared__ float Ft[32 * FS];         // relu'd mid (~66 KB)
  int row0 = blockIdx.x * 32;
  int tid = threadIdx.x;

  for (int idx = tid; idx < 32 * EE; idx += 256) {
    int tr = idx >> 7, c = idx & (EE - 1);
    Mt[tr * MS + c] = msg[(row0 + tr) * EE + c];
  }
  __syncthreads();

  int lane = tid & 31, wave = tid >> 5;
  int m = lane & 15, kh = lane >> 4;

  // ---- GEMM1: K=128, 4 N-tiles x 2 M-subtiles per wave --------------------
  v8f acc[4][2];
#pragma unroll
  for (int t = 0; t < 4; ++t) { acc[t][0] = (v8f){}; acc[t][1] = (v8f){}; }
  {
    const float* Ar0 = Mt + m * MS + 2 * kh;
    const float* Ar1 = Ar0 + 16 * MS;
    int nb = wave * 64 + m;
#pragma unroll 4
    for (int k = 0; k < EE; k += 4) {
      v2f a0; a0.x = Ar0[k]; a0.y = Ar0[k + 1];
      v2f a1; a1.x = Ar1[k]; a1.y = Ar1[k + 1];
      const float* Wk = W1 + (size_t)(k + 2 * kh) * FFD + nb;
#pragma unroll
      for (int t = 0; t < 4; ++t) {
        v2f bf; bf.x = Wk[t * 16]; bf.y = Wk[FFD + t * 16];
        acc[t][0] = wmma4(a0, bf, acc[t][0]);
        acc[t][1] = wmma4(a1, bf, acc[t][1]);
      }
    }
  }
#pragma unroll
  for (int t = 0; t < 4; ++t) {
    int ncol = wave * 64 + t * 16 + m;
    float bi = b1[ncol];
#pragma unroll
    for (int v = 0; v < 8; ++v) {
      float v0 = acc[t][0][v] + bi;
      float v1 = acc[t][1][v] + bi;
      Ft[(kh * 8 + v) * FS + ncol]        = v0 > 0.f ? v0 : 0.f;
      Ft[(16 + kh * 8 + v) * FS + ncol]   = v1 > 0.f ? v1 : 0.f;
    }
  }
  __syncthreads();

  // ---- GEMM2: K=512, one 16-col N tile x 2 M-subtiles per wave ------------
  v8f acc2a = {}, acc2b = {};
  int ncol = wave * 16 + m;
  {
    const float* Ar0 = Ft + m * FS + 2 * kh;
    const float* Ar1 = Ar0 + 16 * FS;
    const float* Wcol = W2 + ncol + 2 * kh * EE;
#pragma unroll 8
    for (int k = 0; k < FFD; k += 4) {
      v2f bf; bf.x = Wcol[k * EE]; bf.y = Wcol[k * EE + EE];
      v2f a0; a0.x = Ar0[k]; a0.y = Ar0[k + 1];
      v2f a1; a1.x = Ar1[k]; a1.y = Ar1[k + 1];
      acc2a = wmma4(a0, bf, acc2a);
      acc2b = wmma4(a1, bf, acc2b);
    }
  }
  // ---- epilogue: residual + bias -----------------------------------------
  float bi = b2[ncol];
  int rb = row0 + kh * 8;
#pragma unroll
  for (int v = 0; v < 8; ++v) {
    hout[(rb + v) * EE + ncol] =
        acc2a[v] + bi + Mt[(kh * 8 + v) * MS + ncol];
    hout[(rb + 16 + v) * EE + ncol] =
        acc2b[v] + bi + Mt[(16 + kh * 8 + v) * MS + ncol];
  }
}

// --------------------------------------------------------------------------
// out[0 : B*J*E]      = h[b, j*MM + (finished ? MM-1 : next_op)]
// out[B*J*E : +HTOT]  = h (flat copy)
// --------------------------------------------------------------------------
__global__ __launch_bounds__(256) void finalize_kernel(
    const float* __restrict__ h, const int* __restrict__ next_op,
    const int* __restrict__ finished, float* __restrict__ out) {
  int idx = blockIdx.x * 256 + threadIdx.x;
  if (idx >= OUTTOT) return;
  if (idx < JOBOUT) {
    int e = idx & (EE - 1);
    int bj = idx >> 7;                  // b*JJ + j
    int j = bj % JJ;
    int b = bj / JJ;
    int ni = (finished[bj] != 0) ? (MM - 1) : next_op[bj];
    out[idx] = h[((b * NND) + j * MM + ni) * EE + e];
  } else {
    out[idx] = h[idx - JOBOUT];
  }
}

// --------------------------------------------------------------------------
extern "C" void kernel_launch(void* const* d_in, const int* in_sizes, int n_in,
                              void* d_out, int out_size, void* d_ws,
                              size_t ws_size, hipStream_t stream) {
  const float* proc     = (const float*)d_in[0];   // (B,J,M) f32
  const int*   mo       = (const int*)d_in[1];     // (B,J,M) i32
  const int*   next_op  = (const int*)d_in[2];     // (B,J)   i32
  const int*   finished = (const int*)d_in[3];     // (B,J)   bool->i32
  const float* W_init   = (const float*)d_in[4];   // (1,E)
  const float* b_init   = (const float*)d_in[5];   // (E)

  // workspace: h ping-pong + machine sums + msg  (~99 MB of f32)
  float* hA  = (float*)d_ws;
  float* hB  = hA + (size_t)HTOT;
  float* Sb  = hB + (size_t)HTOT;
  float* msg = Sb + (size_t)BB * MM * EE;

  init_h_kernel<<<HTOT / 256, 256, 0, stream>>>(proc, W_init, b_init, hA);

  float* hc = hA;
  float* hn = hB;
  for (int l = 0; l < 3; ++l) {
    const int base = 6 + 8 * l;  // sorted-key leaf order per layer dict
    const float* W1 = (const float*)d_in[base + 0];
    const float* W2 = (const float*)d_in[base + 1];
    const float* Wm = (const float*)d_in[base + 2];
    const float* Wp = (const float*)d_in[base + 3];
    const float* Ws = (const float*)d_in[base + 4];
    const float* bv = (const float*)d_in[base + 5];
    const float* b1 = (const float*)d_in[base + 6];
    const float* b2 = (const float*)d_in[base + 7];

    machine_sums_kernel<<<BB, 512, 0, stream>>>(hc, mo, Sb);
    gemm_msg_kernel<<<TILES32, 256, 0, stream>>>(hc, Sb, mo, Ws, Wp, Wm, bv,
                                                 msg);
    ffn_kernel<<<TILES32, 256, 0, stream>>>(msg, W1, b1, W2, b2, hn);
    float* t = hc; hc = hn; hn = t;
  }

  finalize_kernel<<<(OUTTOT + 255) / 256, 256, 0, stream>>>(
      hc, next_op, finished, (float*)d_out);
}